// GCNBlock_1254130450890
// MI455X (gfx1250) — compile-verified
//
#include <hip/hip_runtime.h>

// GCN (3 layers) for MI455X / gfx1250.
// GEMMs via V_WMMA_F32_16X16X4_F32 (fp32 in, fp32 acc, full precision).
// Edge aggregation via global_atomic_add_f32 (feature matrix fits in 192MB L2).

typedef float v2f __attribute__((ext_vector_type(2)));
typedef float v8f __attribute__((ext_vector_type(8)));

#define NN   100000
#define EE   1600000
#define INF_ 128
#define OUTF 64

// ---------------- degree / normalization ----------------

__global__ void k_fill_deg(float* __restrict__ deg) {
    int i = blockIdx.x * 256 + threadIdx.x;
    if (i < NN) deg[i] = 1.0f;               // self-loop contributes 1
}

__global__ void k_count_deg(const int* __restrict__ dst, float* __restrict__ deg) {
    int e = blockIdx.x * 256 + threadIdx.x;
    if (e < EE) atomicAdd(&deg[dst[e]], 1.0f);
}

__global__ void k_rsqrt(float* __restrict__ deg) {
    int i = blockIdx.x * 256 + threadIdx.x;
    if (i < NN) deg[i] = rsqrtf(deg[i]);     // deg >= 1 always (self-loop)
}

// ---------------- WMMA GEMM: H[N,64] = X[N,K] @ W[K,64] ----------------
// One wave -> 16 rows x 64 cols (4 x 16x16 tiles), K-loop in steps of 4.
// A 16x4 f32 layout: lanes 0-15 {K=0,K=1}, lanes 16-31 {K=2,K=3}.
// B 4x16 f32 layout: lane (l&15) = column N, halves select K pairs (mirror of A).

__global__ __launch_bounds__(256)
void k_gemm_wmma(const float* __restrict__ X, const float* __restrict__ W,
                 float* __restrict__ H, int K) {
    const int lane = threadIdx.x & 31;
    const int wave = threadIdx.x >> 5;
    const int half = lane >> 4;     // 0: K-pair (0,1); 1: K-pair (2,3)
    const int mr   = lane & 15;     // row (A) / column (B,D) within tile
    const int m0   = (blockIdx.x * 8 + wave) * 16;
    if (m0 >= NN) return;           // uniform per wave: safe early-out

    v8f acc0 = {}, acc1 = {}, acc2 = {}, acc3 = {};

    const int arow = min(m0 + mr, NN - 1);          // clamp: full EXEC for WMMA
    const float* __restrict__ xrow = X + (size_t)arow * K;

    for (int k = 0; k < K; k += 4) {
        v2f a;
        a.x = xrow[k + 2 * half];
        a.y = xrow[k + 2 * half + 1];

        const float* __restrict__ w0 = W + (size_t)(k + 2 * half) * OUTF + mr;
        const float* __restrict__ w1 = w0 + OUTF;   // next K row
        v2f b0; b0.x = w0[0];  b0.y = w1[0];
        v2f b1; b1.x = w0[16]; b1.y = w1[16];
        v2f b2; b2.x = w0[32]; b2.y = w1[32];
        v2f b3; b3.x = w0[48]; b3.y = w1[48];

        acc0 = __builtin_amdgcn_wmma_f32_16x16x4_f32(false, a, false, b0, (short)0, acc0, false, false);
        acc1 = __builtin_amdgcn_wmma_f32_16x16x4_f32(false, a, false, b1, (short)0, acc1, false, false);
        acc2 = __builtin_amdgcn_wmma_f32_16x16x4_f32(false, a, false, b2, (short)0, acc2, false, false);
        acc3 = __builtin_amdgcn_wmma_f32_16x16x4_f32(false, a, false, b3, (short)0, acc3, false, false);
    }

    // D layout: VGPR v, lanes 0-15 -> (M=v, N=mr); lanes 16-31 -> (M=8+v, N=mr)
#pragma unroll
    for (int v = 0; v < 8; ++v) {
        int r = m0 + half * 8 + v;
        if (r < NN) {
            float* __restrict__ o = H + (size_t)r * OUTF + mr;
            o[0]  = acc0[v];
            o[16] = acc1[v];
            o[32] = acc2[v];
            o[48] = acc3[v];
        }
    }
}

// ---------------- edge scatter: AGG[dst] += norm * H[src] ----------------
// 16 threads per edge, float4 (4 channels) per thread -> coalesced 256B rows.

__global__ __launch_bounds__(256)
void k_edges(const int* __restrict__ src, const int* __restrict__ dst,
             const float* __restrict__ dinv, const float* __restrict__ H,
             float* __restrict__ AGG) {
    long long gid = (long long)blockIdx.x * 256 + threadIdx.x;
    int e = (int)(gid >> 4);
    if (e >= EE) return;
    int c = (int)(gid & 15) * 4;

    int s = src[e], d = dst[e];
    float w = dinv[s] * dinv[d];

    const float4 h4 = *reinterpret_cast<const float4*>(H + (size_t)s * OUTF + c);
    float* __restrict__ base = AGG + (size_t)d * OUTF + c;
    atomicAdd(base + 0, w * h4.x);
    atomicAdd(base + 1, w * h4.y);
    atomicAdd(base + 2, w * h4.z);
    atomicAdd(base + 3, w * h4.w);
}

// ---------------- epilogue: self-loop + bias (+relu) (+residual) ----------------

__global__ __launch_bounds__(256)
void k_epilogue(const float* __restrict__ AGG, const float* __restrict__ H,
                const float* __restrict__ dinv, const float* __restrict__ b,
                const float* __restrict__ res, float* __restrict__ out,
                int do_relu) {
    long long gid = (long long)blockIdx.x * 256 + threadIdx.x;
    if (gid >= (long long)NN * OUTF) return;
    int i = (int)(gid >> 6);
    int c = (int)(gid & 63);
    float di = dinv[i];
    float v = AGG[gid] + di * di * H[gid] + b[c];   // self-loop norm = dinv^2
    if (do_relu) v = fmaxf(v, 0.0f);
    if (res)     v += res[gid];
    out[gid] = v;
}

// ---------------- launcher ----------------

extern "C" void kernel_launch(void* const* d_in, const int* in_sizes, int n_in,
                              void* d_out, int out_size, void* d_ws, size_t ws_size,
                              hipStream_t stream) {
    const float* x    = (const float*)d_in[0];             // [N,128]
    const int*   edge = (const int*)d_in[1];               // [2,E] row-major
    const float* W0   = (const float*)d_in[2];             // [128,64]
    const float* b0   = (const float*)d_in[3];             // [64]
    const float* Ws   = (const float*)d_in[4];             // [2,64,64]
    const float* bs   = (const float*)d_in[5];             // [2,64]
    float*       out  = (float*)d_out;                     // [N,64]

    const int* src = edge;
    const int* dst = edge + EE;

    // workspace: dinv[N] | H[N*64] | AGG[N*64] | XTEMP[N*64]  (~77 MB)
    float* ws   = (float*)d_ws;
    float* dinv = ws;
    float* H    = ws + NN;
    float* AGG  = H  + (size_t)NN * OUTF;
    float* XT   = AGG + (size_t)NN * OUTF;

    const int nBlocks   = (NN + 255) / 256;
    const int eBlocks   = (EE + 255) / 256;
    const int gemmGrid  = (NN + 127) / 128;                // 8 waves x 16 rows / block
    const int edgeGrid  = (int)(((long long)EE * 16 + 255) / 256);
    const int epiGrid   = (int)(((long long)NN * OUTF + 255) / 256);
    const size_t aggBytes = (size_t)NN * OUTF * sizeof(float);

    // symmetric GCN normalization: deg over A+I, dinv = deg^-1/2
    k_fill_deg <<<nBlocks, 256, 0, stream>>>(dinv);
    k_count_deg<<<eBlocks, 256, 0, stream>>>(dst, dinv);
    k_rsqrt    <<<nBlocks, 256, 0, stream>>>(dinv);

    // ---- layer 0: x_temp = conv0(x) ----
    hipMemsetAsync(AGG, 0, aggBytes, stream);
    k_gemm_wmma<<<gemmGrid, 256, 0, stream>>>(x, W0, H, INF_);
    k_edges    <<<edgeGrid, 256, 0, stream>>>(src, dst, dinv, H, AGG);
    k_epilogue <<<epiGrid,  256, 0, stream>>>(AGG, H, dinv, b0, nullptr, XT, 0);

    // ---- layer 1: h = relu(conv1(x_temp)) -> d_out ----
    hipMemsetAsync(AGG, 0, aggBytes, stream);
    k_gemm_wmma<<<gemmGrid, 256, 0, stream>>>(XT, Ws, H, OUTF);
    k_edges    <<<edgeGrid, 256, 0, stream>>>(src, dst, dinv, H, AGG);
    k_epilogue <<<epiGrid,  256, 0, stream>>>(AGG, H, dinv, bs, nullptr, out, 1);

    // ---- layer 2: out = relu(conv2(h)) + x_temp ----
    hipMemsetAsync(AGG, 0, aggBytes, stream);
    k_gemm_wmma<<<gemmGrid, 256, 0, stream>>>(out, Ws + OUTF * OUTF, H, OUTF);
    k_edges    <<<edgeGrid, 256, 0, stream>>>(src, dst, dinv, H, AGG);
    k_epilogue <<<epiGrid,  256, 0, stream>>>(AGG, H, dinv, bs + OUTF, XT, out, 1);
}